// NormalizeAggregator_35639638622225
// MI455X (gfx1250) — compile-verified
//
#include <hip/hip_runtime.h>

typedef __attribute__((ext_vector_type(2))) float v2f;
typedef __attribute__((ext_vector_type(4))) float v4f;
typedef __attribute__((ext_vector_type(8))) float v8f;

#define KDIM   16
#define DDIM   128
#define EDIM   8
#define HALF_D 64
#define AG_STRIDE 132   // 128 + 4 pad: row stride 528B (16B aligned), ≡4 mod 64 banks

__global__ __launch_bounds__(128)
void NormalizeAggregator_kernel(const float* __restrict__ msg,
                                const int*   __restrict__ e_type,
                                const float* __restrict__ e_count,
                                const float* __restrict__ W1,
                                const float* __restrict__ b1,
                                const float* __restrict__ W2,
                                const float* __restrict__ b2,
                                float* __restrict__ out)
{
    __shared__ float s_w[16][16];          // per-(n,k) ratio weight
    __shared__ float s_invtot[16];         // 1 / sum_e e_count0
    __shared__ float s_agg[2 * 16 * AG_STRIDE]; // [0]=nei_msg, [1]=norm_nei_msg

    const int t    = threadIdx.x;          // 0..127
    const int wave = t >> 5;               // 0..3
    const int lane = t & 31;
    const long long n_base = (long long)blockIdx.x * 16;

    // ---- phase 1: tiny per-row weights (reads only the k==0 slice of e_count)
    for (int idx = t; idx < 256; idx += 128) {
        const int nl = idx >> 4, k = idx & 15;
        const long long n = n_base + nl;
        const int et = e_type[n * KDIM + k];
        s_w[nl][k] = 1.0f / e_count[n * (KDIM * EDIM) + et];
    }
    if (t < 16) {
        const long long n = n_base + t;
        float s = 0.f;
#pragma unroll
        for (int e = 0; e < EDIM; ++e) s += e_count[n * (KDIM * EDIM) + e];
        s_invtot[t] = 1.0f / s;
    }
    __syncthreads();

    // ---- phase 2: aggregation. One wave per row, lane q owns 4 columns:
    // global_load_b128, 512B contiguous per wave request, NT hint (stream once,
    // don't evict W1/W2 from WGP$/L2).
    const int d0 = lane * 4;
    for (int nl0 = 0; nl0 < 16; nl0 += 4) {
        const int nl = nl0 + wave;
        const v4f* p = (const v4f*)(msg + (n_base + nl) * (KDIM * DDIM) + d0);
        v4f s1 = {0.f, 0.f, 0.f, 0.f};
        v4f s2 = {0.f, 0.f, 0.f, 0.f};
#pragma unroll
        for (int k = 0; k < KDIM; ++k) {
            const v4f v = __builtin_nontemporal_load(p + k * (DDIM / 4));
            const float w = s_w[nl][k];
            s1 += w * v;
            s2 += v;
        }
        *(v4f*)(&s_agg[nl * AG_STRIDE + d0])                  = s1;                 // nei_msg
        *(v4f*)(&s_agg[16 * AG_STRIDE + nl * AG_STRIDE + d0]) = s2 * s_invtot[nl];  // norm_nei_msg
    }
    __syncthreads();

    // ---- phase 3: full-precision WMMA GEMM  agg(16x128) @ W(128x64) + b
    const int half = lane >> 4;     // 0: lanes 0-15, 1: lanes 16-31
    const int m    = lane & 15;

#pragma unroll
    for (int which = 0; which < 2; ++which) {
        const int tile = wave + which * 4;               // 0..3 -> nei/W1, 4..7 -> norm/W2
        const int src_off = (tile < 4) ? 0 : 16 * AG_STRIDE;
        const float* __restrict__ W    = (tile < 4) ? W1 : W2;
        const float* __restrict__ bias = (tile < 4) ? b1 : b2;
        const int col0 = (tile & 3) * 16;

        // C init: bias depends only on column N = m
        const float bval = bias[col0 + m];
        v8f c;
#pragma unroll
        for (int i = 0; i < 8; ++i) c[i] = bval;

        for (int k0 = 0; k0 < DDIM; k0 += 4) {
            const int ka = k0 + 2 * half;    // lanes 0-15: K={k0,k0+1}; 16-31: K={k0+2,k0+3}
            v2f a, bf;
            a.x  = s_agg[src_off + m * AG_STRIDE + ka];
            a.y  = s_agg[src_off + m * AG_STRIDE + ka + 1];
            bf.x = W[ka * HALF_D + col0 + m];        // B[K][N], row K striped over lanes (L2/WGP$ hit)
            bf.y = W[(ka + 1) * HALF_D + col0 + m];
            c = __builtin_amdgcn_wmma_f32_16x16x4_f32(
                    /*neg_a=*/false, a, /*neg_b=*/false, bf,
                    /*c_mod=*/(short)0, c, /*reuse_a=*/false, /*reuse_b=*/false);
        }

        const int out_col0 = tile * 16;
#pragma unroll
        for (int v = 0; v < 8; ++v) {
            const int mrow = v + 8 * half;   // C layout: VGPR v -> M=v (half 0) / v+8 (half 1)
            __builtin_nontemporal_store(c[v], &out[(n_base + mrow) * DDIM + out_col0 + m]);
        }
    }
}

extern "C" void kernel_launch(void* const* d_in, const int* in_sizes, int n_in,
                              void* d_out, int out_size, void* d_ws, size_t ws_size,
                              hipStream_t stream) {
    (void)n_in; (void)d_ws; (void)ws_size; (void)out_size;
    const float* msg     = (const float*)d_in[0];
    const int*   e_type  = (const int*)  d_in[1];
    const float* e_count = (const float*)d_in[2];
    const float* W1      = (const float*)d_in[3];
    const float* b1      = (const float*)d_in[4];
    const float* W2      = (const float*)d_in[5];
    const float* b2      = (const float*)d_in[6];
    float* out = (float*)d_out;

    const long long N = (long long)in_sizes[0] / (KDIM * DDIM);  // 100000
    const int nblocks = (int)(N / 16);                            // 6250 (N % 16 == 0)

    NormalizeAggregator_kernel<<<nblocks, 128, 0, stream>>>(
        msg, e_type, e_count, W1, b1, W2, b2, out);
}